// Temporal_Operator_33182917329257
// MI455X (gfx1250) — compile-verified
//
#include <hip/hip_runtime.h>

// STL Always_[16,64]: out[t] = min(signal[t-63 .. t-15]), 1e6 padding, per column.
// Bandwidth-bound (256 MiB compulsory -> ~11.5 us @ 23.3 TB/s). Gil-Werman block
// scan = ~3 v_min_num_f32 per element. MI455X-specific path: Tensor Data Mover
// bulk-loads each workgroup's 147-row x 256-col fp32 tile (147 KB, 2 WGs/WGP in
// the 320 KB LDS) with ONE tensor_load_to_lds, tracked by TENSORcnt; threads
// read conflict-free ds_load_b32 columns; outputs leave via non-temporal stores
// so the 192 MB L2 stays reserved for input halo re-reads.

typedef unsigned int u32;
typedef unsigned int u32x4 __attribute__((ext_vector_type(4)));
typedef int          i32x4 __attribute__((ext_vector_type(4)));
typedef int          i32x8 __attribute__((ext_vector_type(8)));

namespace {
constexpr int   T_DIM  = 8192;           // time steps
constexpr int   B_DIM  = 4096;           // batch columns (fastest dim)
constexpr int   WIN    = 49;             // STEPS
constexpr int   DELAY  = 15;             // out[t] = rmin(t-15)
constexpr int   TILE   = 2 * WIN;        // 98 outputs per thread
constexpr int   ROWS   = TILE + WIN;     // 147 fetched rows (incl. halo)
constexpr int   NC     = 256;            // columns per workgroup (8 wave32s)
constexpr int   NJ     = T_DIM - DELAY;  // valid window positions: 0..8176
constexpr int   NTILES = (NJ + TILE - 1) / TILE;   // 84
constexpr float PAD    = 1.0e6f;         // reference LARGE, exact
}

__global__ __launch_bounds__(NC)
void Temporal_Operator_tdm_slidemin(const float* __restrict__ sig,
                                    float* __restrict__ out)
{
    __shared__ float lbuf[ROWS * NC];                 // 147 KB static LDS

    const int tid   = threadIdx.x;
    const int b0    = blockIdx.x * NC;
    const int b     = b0 + tid;
    const int p     = blockIdx.y;
    const int jbase = p * TILE;
    const int row0  = jbase - WIN;                    // first halo row (may be < 0)
    const int fstart = (row0 < 0) ? 0 : row0;
    const int miss   = fstart - row0;                 // 49 for p==0, else 0
    int nrows = ROWS - miss;
    if (fstart + nrows > T_DIM) nrows = T_DIM - fstart;

    // Leading rows t = 0..14 are exactly LARGE in the reference.
    if (p == 0) {
#pragma unroll
        for (int t = 0; t < DELAY; ++t)
            __builtin_nontemporal_store(PAD, &out[(size_t)t * B_DIM + b]);
    }

#if __has_builtin(__builtin_amdgcn_tensor_load_to_lds) && \
    __has_builtin(__builtin_amdgcn_s_wait_tensorcnt)
    // ---- TDM: one DMA descriptor fetches the whole tile into LDS ----
    if (tid < 32) {   // wave 0 issues; EXEC is ignored by TDM, TENSORcnt is per-wave
        const unsigned long long gaddr =
            (unsigned long long)(const void*)sig +
            4ull * ((unsigned long long)fstart * B_DIM + (unsigned long long)b0);
        // Generic pointer to LDS: low 32 bits are the raw LDS byte address.
        const u32 laddr =
            (u32)(unsigned long long)(const void*)&lbuf[0] + (u32)miss * (NC * 4u);

        u32x4 g0;                       // D# group 0 (ISA 8.3)
        g0[0] = 1u;                                        // count=1, user desc
        g0[1] = laddr;                                     // lds_addr (bytes)
        g0[2] = (u32)gaddr;                                // global_addr[31:0]
        g0[3] = ((u32)(gaddr >> 32) & 0x01FFFFFFu)         // global_addr[56:32]
                | (2u << 30);                              // type = 2 ("image")

        i32x8 g1;                       // D# group 1 (ISA 8.4)
        g1[0] = (int)(2u << 16);                           // data_size = 4 bytes
        g1[1] = (int)((u32)B_DIM << 16);                   // tensor_dim0[15:0]
        g1[2] = (int)(((u32)B_DIM >> 16)                   // tensor_dim0[31:16]
                      | ((u32)T_DIM << 16));               // tensor_dim1[15:0]
        g1[3] = (int)(((u32)T_DIM >> 16)                   // tensor_dim1[31:16]
                      | ((u32)NC << 16));                  // tile_dim0 = 256 cols
        g1[4] = (int)((u32)nrows & 0xFFFFu);               // tile_dim1 = rows, tile_dim2=0
        g1[5] = (int)B_DIM;                                // tensor_dim0_stride lo32 (elements)
        g1[6] = 0;                                         // stride hi / dim1_stride lo
        g1[7] = 0;
        i32x4 g2 = {0, 0, 0, 0};        // 2D tensor: groups 2/3 unused
        i32x4 g3 = {0, 0, 0, 0};
        i32x8 g4 = {0, 0, 0, 0, 0, 0, 0, 0};   // extra operand (clang-23 6-arg form)

        __builtin_amdgcn_tensor_load_to_lds(g0, g1, g2, g3, g4, 0);
        __builtin_amdgcn_s_wait_tensorcnt(0);
    }
#else
    // ---- Fallback: cooperative staging with plain coalesced loads ----
#pragma unroll 4
    for (int i = 0; i < ROWS; ++i) {
        const int g = row0 + i;
        if (g >= 0 && g < T_DIM)
            lbuf[i * NC + tid] = sig[(size_t)g * B_DIM + b];
    }
#endif
    __syncthreads();

    const float INF = __builtin_inff();
    float prevS[WIN];   // suffix mins of previous 49-block
    float cur[WIN];     // raw values of current block

    // Halo block (LDS rows 0..48); rows before t=0 are PAD (match reference).
#pragma unroll
    for (int i = 0; i < WIN; ++i)
        cur[i] = (row0 + i >= 0) ? lbuf[i * NC + tid] : PAD;
    prevS[WIN - 1] = cur[WIN - 1];
#pragma unroll
    for (int i = WIN - 2; i >= 0; --i)
        prevS[i] = fminf(cur[i], prevS[i + 1]);

#pragma unroll
    for (int blk = 0; blk < 2; ++blk) {
        const int base = jbase + blk * WIN;
        float run = INF;
#pragma unroll
        for (int i = 0; i < WIN; ++i) {
            const int j = base + i;
            const float v =
                (j < T_DIM) ? lbuf[(WIN + blk * WIN + i) * NC + tid] : PAD;
            cur[i] = v;
            run = fminf(run, v);
            const float o = fminf((i + 1 < WIN) ? prevS[i + 1] : INF, run);
            if (j < NJ)   // j <= 8176  ->  t = j+15 <= 8191
                __builtin_nontemporal_store(o, &out[(size_t)(j + DELAY) * B_DIM + b]);
        }
        if (blk == 0) {   // current block becomes next block's suffix
            prevS[WIN - 1] = cur[WIN - 1];
#pragma unroll
            for (int i = WIN - 2; i >= 0; --i)
                prevS[i] = fminf(cur[i], prevS[i + 1]);
        }
    }
}

extern "C" void kernel_launch(void* const* d_in, const int* in_sizes, int n_in,
                              void* d_out, int out_size, void* d_ws, size_t ws_size,
                              hipStream_t stream) {
    (void)in_sizes; (void)n_in; (void)out_size; (void)d_ws; (void)ws_size;
    const float* sig = (const float*)d_in[0];
    float*       out = (float*)d_out;

    dim3 block(NC);
    dim3 grid(B_DIM / NC, NTILES);    // 16 x 84 workgroups, 2 per WGP (147 KB LDS)
    Temporal_Operator_tdm_slidemin<<<grid, block, 0, stream>>>(sig, out);
}